// CoreProcessor_53437983097426
// MI455X (gfx1250) — compile-verified
//
#include <hip/hip_runtime.h>
#include <stdint.h>

// ---------------------------------------------------------------------------
// CoreProcessor for MI455X (gfx1250, wave32)
//   processed[t,:] = sum_m attn[t,m] * (ops[m] @ x[t])
//                  = sum_m ( (attn[t,m]*x[t]) @ ops[m]^T )   <-- fold scale into A rows
// Main contraction: 34.4 GFLOP f32 -> V_WMMA_F32_16X16X4_F32, K chained over m*e.
// ops[m] streamed HBM/L2 -> LDS with the Tensor Data Mover (double buffered),
// using TDM pad_enable to write a 132-float pitch (pitch%64==4 => ds_load_b64
// B fetches are bank-conflict free).
// ---------------------------------------------------------------------------

#define DIMD   128
#define MSLOTS 64
#define NTOK   16384          // B*S = 4*4096
#define TOK_PER_WG 64
#define PITCH  132            // LDS row pitch in floats (128 + 4 pad)
#define NTHREADS 128          // 4 waves

typedef float v2f  __attribute__((ext_vector_type(2)));
typedef float v8f  __attribute__((ext_vector_type(8)));
typedef unsigned int v4u __attribute__((ext_vector_type(4)));
typedef int   v4i  __attribute__((ext_vector_type(4)));
typedef int   v8i  __attribute__((ext_vector_type(8)));

// LDS layout (floats)
#define XS_OFF   0
#define ATTN_OFF (64 * PITCH)                 // attnT[64][64]
#define QS_OFF   (ATTN_OFF + 64 * 64)
#define OPS_OFF  (QS_OFF + 64 * PITCH)
#define OPS_BUF_FLOATS (128 * PITCH)
#define SMEM_FLOATS (OPS_OFF + 2 * OPS_BUF_FLOATS)   // 54784 floats = 219136 B

__device__ __forceinline__ unsigned lds_off_of(const void* p) {
  // generic pointers to LDS: addr[31:0] is the LDS byte offset (ISA aperture rule)
  return (unsigned)(uintptr_t)p;
}

// TDM: load one 128x128 f32 tile (ops[m]) to LDS with 16B padding per 512B row.
__device__ __forceinline__ void tdm_load_ops_tile(const float* gsrc, unsigned lds_byte_off) {
  uint64_t ga = (uint64_t)(uintptr_t)gsrc;
  v4u g0;
  g0[0] = 1u;                                        // count=1, user descriptor
  g0[1] = lds_byte_off;                              // lds_addr
  g0[2] = (unsigned)(ga & 0xFFFFFFFFull);            // global_addr[31:0]
  g0[3] = (unsigned)((ga >> 32) & 0x01FFFFFFull)     // global_addr[56:32]
        | (2u << 30);                                // type = 2 ("image")
  v8i g1;
  g1[0] = (int)((2u << 16)      // data_size = 4B
              | (1u << 20)      // pad_enable
              | (6u << 22)      // pad_interval code 6 => every 128 dwords
              | (3u << 25));    // pad_amount  code 3 => 4 dwords (pitch 132)
  g1[1] = (int)(128u << 16);                 // tensor_dim0 = 128      (bits 79:48)
  g1[2] = (int)((8192u & 0xFFFFu) << 16);    // tensor_dim1 = 64*128   (bits 111:80)
  g1[3] = (int)(128u << 16);                 // tile_dim0 = 128        (bits 127:112)
  g1[4] = (int)128;                          // tile_dim1 = 128, tile_dim2 = 0
  g1[5] = (int)128;                          // tensor_dim0_stride = 128 (low32)
  g1[6] = 0;
  g1[7] = 0;
  v4i z4 = {0, 0, 0, 0};
  v8i z8 = {0, 0, 0, 0, 0, 0, 0, 0};
  // 6-arg form (clang-23 / therock-10.0 headers): groups 0..3 + extra group + cpol
  __builtin_amdgcn_tensor_load_to_lds(g0, g1, z4, z4, z8, 0);
}

__global__ __launch_bounds__(NTHREADS) void core_processor_kernel(
    const float* __restrict__ x,     // [16384,128]
    const float* __restrict__ keys,  // [64,128]
    const float* __restrict__ ops,   // [64,128,128]
    const float* __restrict__ Wq,    // [128,128]
    const float* __restrict__ bq,    // [128]
    float* __restrict__ out)         // [16384,128]
{
  extern __shared__ float smem[];
  float* xs    = smem + XS_OFF;
  float* attnT = smem + ATTN_OFF;
  float* qs    = smem + QS_OFF;
  float* opsb  = smem + OPS_OFF;

  const int tid  = threadIdx.x;
  const int wave = tid >> 5;
  const int lane = tid & 31;
  const int lrow = lane & 15;       // token row within wave's 16x16 tiles / B column
  const int half = lane >> 4;       // K-half split of WMMA operands
  const int tok_base = blockIdx.x * TOK_PER_WG;

  // Kick off TDM for ops[0] immediately; stage-1 compute hides its latency.
  if (wave == 0) {
    tdm_load_ops_tile(ops, lds_off_of(opsb));
  }

  // ---------------- Stage 1a: stage x tile (64 tokens x 128) into LDS --------
  for (int i = tid; i < 64 * 32; i += NTHREADS) {     // float4 granularity
    int r  = i >> 5;
    int c4 = i & 31;
    float4 v = *(const float4*)(x + (size_t)(tok_base + r) * DIMD + c4 * 4);
    *(float4*)(xs + r * PITCH + c4 * 4) = v;
  }
  __syncthreads();

  // ---------------- Stage 1b: queries = x @ Wq^T + bq ------------------------
  {
    int t     = tid >> 1;
    int dbase = (tid & 1) * 64;
    for (int i = 0; i < 64; ++i) {
      int d = dbase + i;
      const float4* wrow = (const float4*)(Wq + (size_t)d * DIMD);
      float s = bq[d];
      for (int e4 = 0; e4 < 32; ++e4) {
        float4 w  = wrow[e4];
        float4 xv = *(const float4*)(xs + t * PITCH + e4 * 4);
        s += w.x * xv.x + w.y * xv.y + w.z * xv.z + w.w * xv.w;
      }
      qs[t * PITCH + d] = s;
    }
  }
  __syncthreads();

  // ---------------- Stage 1c: logits + softmax -> attnT[m][t] ---------------
  if (tid < 64) {
    int t = tid;
    const float scale = 0.08838834764831845f;   // 1/sqrt(128)
    float mx = -3.0e38f;
    for (int m = 0; m < MSLOTS; ++m) {
      const float4* krow = (const float4*)(keys + (size_t)m * DIMD);
      float s = 0.f;
      for (int e4 = 0; e4 < 32; ++e4) {
        float4 k  = krow[e4];
        float4 qv = *(const float4*)(qs + t * PITCH + e4 * 4);
        s += k.x * qv.x + k.y * qv.y + k.z * qv.z + k.w * qv.w;
      }
      s *= scale;
      attnT[m * 64 + t] = s;
      mx = fmaxf(mx, s);
    }
    float sum = 0.f;
    for (int m = 0; m < MSLOTS; ++m) {
      float e = __expf(attnT[m * 64 + t] - mx);
      attnT[m * 64 + t] = e;
      sum += e;
    }
    float inv = 1.f / sum;
    for (int m = 0; m < MSLOTS; ++m) attnT[m * 64 + t] *= inv;
  }
  __syncthreads();

  // ---------------- Stage 2: main WMMA contraction ---------------------------
  // A registers: lane holds token row (wave*16+lrow); K-groups of 4 split by half.
  v2f xa[32];
  {
    const float* xrowp = xs + (wave * 16 + lrow) * PITCH + 2 * half;
    #pragma unroll
    for (int g = 0; g < 32; ++g) xa[g] = *(const v2f*)(xrowp + 4 * g);
  }
  v8f acc[8] = {};   // 16 tokens x 128 d, as 8 16x16 f32 C tiles

  for (int m = 0; m < MSLOTS; ++m) {
    if (wave == 0) {
      if (m + 1 < MSLOTS) {
        tdm_load_ops_tile(ops + (size_t)(m + 1) * DIMD * DIMD,
                          lds_off_of(opsb + ((m + 1) & 1) * OPS_BUF_FLOATS));
        __builtin_amdgcn_s_wait_tensorcnt(1);   // tile m is complete (in-order)
      } else {
        __builtin_amdgcn_s_wait_tensorcnt(0);
      }
    }
    __syncthreads();   // publish buffer (m&1) to all waves

    // per-m row scale: one scalar per lane (lane == token row in A layout)
    float aw = attnT[m * 64 + (wave * 16 + lrow)];
    v2f xsc[32];
    #pragma unroll
    for (int g = 0; g < 32; ++g) xsc[g] = xa[g] * aw;

    const float* brow = opsb + (m & 1) * OPS_BUF_FLOATS + lrow * PITCH + 2 * half;
    #pragma unroll
    for (int g = 0; g < 32; ++g) {
      #pragma unroll
      for (int dt = 0; dt < 8; ++dt) {   // 8 independent accumulator chains
        v2f bv = *(const v2f*)(brow + dt * 16 * PITCH + 4 * g);
        acc[dt] = __builtin_amdgcn_wmma_f32_16x16x4_f32(
            false, xsc[g], false, bv, (short)0, acc[dt], false, false);
      }
    }
    __syncthreads();   // buffer (m&1) free for TDM overwrite at iteration m+1
  }

  // ---------------- Epilogue: C layout -> global (coalesced 64B segments) ----
  #pragma unroll
  for (int dt = 0; dt < 8; ++dt) {
    #pragma unroll
    for (int j = 0; j < 8; ++j) {
      int row = tok_base + wave * 16 + j + 8 * half;   // C: M = j + 8*half
      out[(size_t)row * DIMD + dt * 16 + lrow] = acc[dt][j];
    }
  }
}

extern "C" void kernel_launch(void* const* d_in, const int* in_sizes, int n_in,
                              void* d_out, int out_size, void* d_ws, size_t ws_size,
                              hipStream_t stream) {
  (void)in_sizes; (void)n_in; (void)out_size; (void)d_ws; (void)ws_size;
  const float* x    = (const float*)d_in[0];
  const float* keys = (const float*)d_in[1];
  const float* ops  = (const float*)d_in[2];
  const float* Wq   = (const float*)d_in[3];
  const float* bq   = (const float*)d_in[4];
  float* out = (float*)d_out;

  const size_t shmem = (size_t)SMEM_FLOATS * sizeof(float);   // 219136 B
  (void)hipFuncSetAttribute((const void*)core_processor_kernel,
                            hipFuncAttributeMaxDynamicSharedMemorySize, (int)shmem);
  dim3 grid(NTOK / TOK_PER_WG);   // 256
  dim3 block(NTHREADS);           // 128 = 4 waves
  core_processor_kernel<<<grid, block, shmem, stream>>>(x, keys, ops, Wq, bq, out);
}